// FastAP_31301721653821
// MI455X (gfx1250) — compile-verified
//
#include <hip/hip_runtime.h>
#include <hip/hip_bf16.h>
#include <math.h>

#define N_ROWS 16384
#define DIMS   128
#define NCLS   100
#define NBINS  2048
#define BIN_LEN (4.0f / 2048.0f)
#define INV_BIN_LEN (2048.0f / 4.0f)
#define EPS_F  1e-7f

typedef __attribute__((ext_vector_type(2))) float v2f;
typedef __attribute__((ext_vector_type(8))) float v8f;

// ---------------- workspace layout (bytes) ----------------
//  [0      ,  8192) : float h[2048]
//  [8192   , 16384) : float h_plus[2048]
//  [16384  , 16388) : float n_plus
//  [16416  , 81952) : int   labels[16384]

__global__ void fastap_init_kernel(float* hist) {
    int i = blockIdx.x * blockDim.x + threadIdx.x;
    if (i < 2 * NBINS + 1) hist[i] = 0.0f;   // h, h_plus, n_plus are contiguous floats
}

// One thread per row: labels[i] = argmax_c Y[i, c] (first max wins, like jnp.argmax)
__global__ void fastap_argmax_kernel(const float* __restrict__ Y, int* __restrict__ labels) {
    int i = blockIdx.x * blockDim.x + threadIdx.x;
    if (i >= N_ROWS) return;
    const float* row = Y + (size_t)i * NCLS;
    float best = row[0];
    int   bi   = 0;
    for (int c = 1; c < NCLS; ++c) {
        float v = row[c];
        if (v > best) { best = v; bi = c; }
    }
    labels[i] = bi;
}

// Each wave (32 lanes) computes 16 distances via V_WMMA_F32_16X16X4_F32:
//   A = (r - q)^2 tile (16 x 4, f32), B = all-ones (4 x 16) -> D[m][*] = d_m.
// Then 16 lanes scatter triangular-kernel weights into global histograms.
__global__ void __launch_bounds__(256)
fastap_dist_hist_kernel(const float* __restrict__ out,
                        const int*   __restrict__ labels,
                        float* __restrict__ h,
                        float* __restrict__ hp,
                        float* __restrict__ n_plus) {
    __shared__ float q[DIMS];
    int tid = threadIdx.x;
    if (tid < DIMS) q[tid] = out[tid];           // query = row 0, staged in LDS
    __syncthreads();

    int lane    = tid & 31;
    int wave    = tid >> 5;
    int rowbase = blockIdx.x * 128 + wave * 16;  // 8 waves * 16 rows = 128 rows/block
    int m       = lane & 15;                     // A-matrix M index for this lane
    int koff    = (lane >> 4) << 1;              // lanes 0-15: K+0/1, lanes 16-31: K+2/3

    const float* rp = out + (size_t)(rowbase + m) * DIMS;

    v8f acc = {};
    v2f bones;
    bones.x = 1.0f; bones.y = 1.0f;
    for (int kb = 0; kb < DIMS; kb += 4) {
        int k = kb + koff;
        float2 r = *(const float2*)(rp + k);     // 8B aligned: rows are 512B aligned, k even
        float d0 = r.x - q[k];
        float d1 = r.y - q[k + 1];
        v2f a;
        a.x = d0 * d0;
        a.y = d1 * d1;
        // D = A * ones + C  -> every column of row m holds d_m
        acc = __builtin_amdgcn_wmma_f32_16x16x4_f32(
            /*neg_a=*/false, a, /*neg_b=*/false, bones,
            /*c_mod=*/(short)0, acc, /*reuse_a=*/false, /*reuse_b=*/false);
    }

    // C/D layout: lanes 0-15 VGPR v = row v; lanes 16-31 VGPR v = row v+8.
    int   row_local = -1;
    float d = 0.0f;
    if (lane < 8)                     { row_local = lane;     d = acc[lane]; }
    else if (lane >= 16 && lane < 24) { row_local = lane - 8; d = acc[lane - 16]; }

    if (row_local >= 0) {
        int  row = rowbase + row_local;
        bool pos = (labels[row] == labels[0]);
        if (pos) atomicAdd(n_plus, 1.0f);        // n_plus counts row 0 too
        if (row != 0) {                          // histogram excludes the query row
            float t  = d * INV_BIN_LEN - 0.5f;   // fractional bin position
            int   ib = (int)floorf(t);
            #pragma unroll
            for (int s = 0; s < 2; ++s) {
                int j = ib + s;
                if (j >= 0 && j < NBINS) {
                    float w = 1.0f - fabsf(((float)j + 0.5f) * BIN_LEN - d) * INV_BIN_LEN;
                    if (w > 0.0f) {
                        atomicAdd(&h[j], w);
                        if (pos) atomicAdd(&hp[j], w);
                    }
                }
            }
        }
    }
}

// Single block: cumsum over 2048 bins (8 bins/thread + LDS scan of 256 partials),
// then loss = sum(h_plus * H_plus / (H + eps)) / n_plus.
__global__ void __launch_bounds__(256)
fastap_finalize_kernel(const float* __restrict__ h,
                       const float* __restrict__ hp,
                       const float* __restrict__ n_plus,
                       float* __restrict__ out) {
    __shared__ float sh[256];
    __shared__ float shp[256];
    int tid  = threadIdx.x;
    int base = tid * 8;

    float vh[8], vhp[8], lh[8], lhp[8];
    float th = 0.0f, thp = 0.0f;
    #pragma unroll
    for (int j = 0; j < 8; ++j) {
        vh[j]  = h[base + j];
        vhp[j] = hp[base + j];
        th  += vh[j];  lh[j]  = th;    // local inclusive prefix
        thp += vhp[j]; lhp[j] = thp;
    }
    sh[tid] = th; shp[tid] = thp;
    __syncthreads();

    // Hillis-Steele inclusive scan of per-thread totals
    for (int off = 1; off < 256; off <<= 1) {
        float a = (tid >= off) ? sh[tid - off]  : 0.0f;
        float b = (tid >= off) ? shp[tid - off] : 0.0f;
        __syncthreads();
        sh[tid]  += a;
        shp[tid] += b;
        __syncthreads();
    }
    float eh  = sh[tid]  - th;    // exclusive prefix for this thread's chunk
    float ehp = shp[tid] - thp;

    float partial = 0.0f;
    #pragma unroll
    for (int j = 0; j < 8; ++j) {
        float H  = eh  + lh[j];
        float Hp = ehp + lhp[j];
        partial += vhp[j] * Hp / (H + EPS_F);
    }

    __syncthreads();
    sh[tid] = partial;
    __syncthreads();
    for (int s = 128; s > 0; s >>= 1) {
        if (tid < s) sh[tid] += sh[tid + s];
        __syncthreads();
    }
    if (tid == 0) out[0] = sh[0] / n_plus[0];
}

extern "C" void kernel_launch(void* const* d_in, const int* in_sizes, int n_in,
                              void* d_out, int out_size, void* d_ws, size_t ws_size,
                              hipStream_t stream) {
    const float* output = (const float*)d_in[0];
    // d_in[1] (pos_output) and d_in[2] (neg_output) are unused by the reference.
    const float* Y      = (const float*)d_in[3];

    char*  ws     = (char*)d_ws;
    float* h      = (float*)(ws);
    float* hplus  = (float*)(ws + 8192);
    float* nplus  = (float*)(ws + 16384);
    int*   labels = (int*)  (ws + 16416);

    fastap_init_kernel<<<(2 * NBINS + 1 + 255) / 256, 256, 0, stream>>>(h);
    fastap_argmax_kernel<<<N_ROWS / 256, 256, 0, stream>>>(Y, labels);
    fastap_dist_hist_kernel<<<N_ROWS / 128, 256, 0, stream>>>(output, labels, h, hplus, nplus);
    fastap_finalize_kernel<<<1, 256, 0, stream>>>(h, hplus, nplus, (float*)d_out);
}